// EnvironmentalAugmentations_49520972923587
// MI455X (gfx1250) — compile-verified
//
#include <hip/hip_runtime.h>

// ---------------------------------------------------------------------------
// Pink-noise IIR scan + mix + normalize for MI455X (gfx1250, wave32).
//
// f[t] = A*f[t-1] + B*w[t]  parallelized over T by chunking (A^2048 ~ 1e-9,
// below f32 eps, so each chunk warms up its carry from zero over 2048 steps).
// Each 16-step block is D(16t x 16ch) = M(16x16, lower-tri const) x W + C,
// via 4 chained V_WMMA_F32_16X16X4_F32; carry folded into C, new carry pulled
// from row M=15 with ds_bpermute.  Loop is peeled into warmup / interior /
// tail phases so the hot interior loop has no clamps and no divergent guards.
// ---------------------------------------------------------------------------

typedef float v2f __attribute__((ext_vector_type(2)));
typedef float v8f __attribute__((ext_vector_type(8)));

#define T_LEN   220500
#define A_C     0.99f
#define B_C     0.01f
#define NOISE_L 0.05f
#define CHUNK   4096
#define NCHUNK  54        // 54*4096 = 221184 >= 220500
#define WARMUP  2048      // 0.99^2048 ~ 1.2e-9 (< f32 eps)
#define WAVES_PER_BLOCK 8

// M[i][j] = B * A^(i-j) for i>=j else 0   (computed per-lane once)
__device__ __forceinline__ float iir_coef(int i, int j) {
    if (i < j) return 0.0f;
    float p = B_C;
    for (int d = 0; d < i - j; ++d) p *= A_C;
    return p;
}

__global__ void pink_init_ws(unsigned int* ws) { ws[0] = 0u; }

__global__ void __launch_bounds__(256)
pink_scan_mix(const float* __restrict__ wf,
              const float* __restrict__ wn,
              float*       __restrict__ out,
              unsigned int* __restrict__ maxbits)
{
    const int lane = threadIdx.x & 31;
    const int wave = blockIdx.x * WAVES_PER_BLOCK + (threadIdx.x >> 5);
    const int chunk = wave >> 4;      // 0..53  (time chunk)
    const int chg   = wave & 15;      // 0..15  (channel group of 16)
    const int hi    = lane >> 4;      // upper/lower half-wave
    const int lo    = lane & 15;
    const int ch    = chg * 16 + lo;

    const float* wrow = wn  + (size_t)ch * T_LEN;
    const float* frow = wf  + (size_t)ch * T_LEN;
    float*       orow = out + (size_t)ch * T_LEN;

    // A-operand (constant IIR matrix), 16x4 f32 per WMMA -> 2 VGPRs/lane.
    // ISA layout: lanes0-15 row M=lo, VGPR0={K=k0+2*hi}, VGPR1={K=k0+2*hi+1}.
    v2f aop[4];
#pragma unroll
    for (int k = 0; k < 4; ++k) {
        const int cx = 4 * k + 2 * hi;
        aop[k].x = iir_coef(lo, cx);
        aop[k].y = iir_coef(lo, cx + 1);
    }
    // C/D layout: VGPR r holds time-row (r + 8*hi); carry weight A^(row+1).
    float cpow[8];
#pragma unroll
    for (int r = 0; r < 8; ++r) {
        float p = A_C;
        for (int d = 0; d < r + 8 * hi; ++d) p *= A_C;
        cpow[r] = p;
    }
    const int shufAddr = (16 + lo) << 2;   // byte index for ds_bpermute source

    const int outStart = chunk * CHUNK;
    const int tEnd     = min(outStart + CHUNK, T_LEN);
    float mx = 0.0f;
    float carry;

    // ---- Phase 1: warmup (no output, no clamps, no guards) ----------------
    if (chunk == 0) {
        carry = wrow[0] * ((1.0f - B_C) / A_C);   // makes f[0] == w[0] exact
    } else {
        carry = 0.0f;
        const float* wp = wrow + (outStart - WARMUP) + 2 * hi;
#pragma unroll 2
        for (int b = 0; b < WARMUP / 16; ++b) {
            v8f acc;
#pragma unroll
            for (int r = 0; r < 8; ++r) acc[r] = cpow[r] * carry;
#pragma unroll
            for (int k = 0; k < 4; ++k) {
                const v2f bv = *reinterpret_cast<const v2f*>(wp + 4 * k);
                acc = __builtin_amdgcn_wmma_f32_16x16x4_f32(
                          false, aop[k], false, bv, (short)0, acc, false, false);
            }
            carry = __builtin_bit_cast(
                float, __builtin_amdgcn_ds_bpermute(
                           shufAddr, __builtin_bit_cast(int, acc[7])));
            wp += 16;
        }
    }

    // ---- Phase 2: interior full blocks (branch-free vector path) ----------
    const int nFull = (tEnd - outStart) >> 4;     // 256 (or 213 for chunk 53)
    {
        const float* wp = wrow + outStart + 2 * hi;
        const float* fp = frow + outStart + 8 * hi;
        float*       op = orow + outStart + 8 * hi;
        for (int b = 0; b < nFull; ++b) {
            v8f acc;
#pragma unroll
            for (int r = 0; r < 8; ++r) acc[r] = cpow[r] * carry;
#pragma unroll
            for (int k = 0; k < 4; ++k) {
                const v2f bv = *reinterpret_cast<const v2f*>(wp + 4 * k);
                acc = __builtin_amdgcn_wmma_f32_16x16x4_f32(
                          false, aop[k], false, bv, (short)0, acc, false, false);
            }
            carry = __builtin_bit_cast(
                float, __builtin_amdgcn_ds_bpermute(
                           shufAddr, __builtin_bit_cast(int, acc[7])));

            // mix + max + store: 8 consecutive samples of this lane's channel
            const float4 f0 = *reinterpret_cast<const float4*>(fp);
            const float4 f1 = *reinterpret_cast<const float4*>(fp + 4);
            float4 m0, m1;
            m0.x = fmaf(NOISE_L, acc[0], f0.x);  m0.y = fmaf(NOISE_L, acc[1], f0.y);
            m0.z = fmaf(NOISE_L, acc[2], f0.z);  m0.w = fmaf(NOISE_L, acc[3], f0.w);
            m1.x = fmaf(NOISE_L, acc[4], f1.x);  m1.y = fmaf(NOISE_L, acc[5], f1.y);
            m1.z = fmaf(NOISE_L, acc[6], f1.z);  m1.w = fmaf(NOISE_L, acc[7], f1.w);
            mx = fmaxf(mx, fmaxf(fmaxf(fabsf(m0.x), fabsf(m0.y)),
                                 fmaxf(fabsf(m0.z), fabsf(m0.w))));
            mx = fmaxf(mx, fmaxf(fmaxf(fabsf(m1.x), fabsf(m1.y)),
                                 fmaxf(fabsf(m1.z), fabsf(m1.w))));
            *reinterpret_cast<float4*>(op)     = m0;
            *reinterpret_cast<float4*>(op + 4) = m1;

            wp += 16; fp += 16; op += 16;
        }
    }

    // ---- Phase 3: at most one partial tail block (chunk 53 only) ----------
    const int t0 = outStart + (nFull << 4);
    if (t0 < tEnd) {
        v8f acc;
#pragma unroll
        for (int r = 0; r < 8; ++r) acc[r] = cpow[r] * carry;
#pragma unroll
        for (int k = 0; k < 4; ++k) {
            int tr = t0 + 4 * k + 2 * hi;
            tr = min(tr, T_LEN - 2);             // clamped: tail only
            const v2f bv = *reinterpret_cast<const v2f*>(wrow + tr);
            acc = __builtin_amdgcn_wmma_f32_16x16x4_f32(
                      false, aop[k], false, bv, (short)0, acc, false, false);
        }
#pragma unroll
        for (int r = 0; r < 8; ++r) {
            const int t = t0 + 8 * hi + r;
            if (t < tEnd) {
                const float m = fmaf(NOISE_L, acc[r], frow[t]);
                mx = fmaxf(mx, fabsf(m));
                orow[t] = m;
            }
        }
    }

    // Wave-level max reduction; |x|>=0 so u32 bit-pattern order == float order.
#pragma unroll
    for (int off = 16; off > 0; off >>= 1)
        mx = fmaxf(mx, __shfl_xor(mx, off, 32));
    if (lane == 0)
        atomicMax(maxbits, __float_as_uint(mx));
}

__global__ void __launch_bounds__(256)
pink_normalize(float* __restrict__ out,
               const unsigned int* __restrict__ maxbits, int n4)
{
    const float maxv  = __uint_as_float(maxbits[0]);
    const float scale = (maxv > 1.0f) ? (1.0f / maxv) : 1.0f;
    float4* o4 = reinterpret_cast<float4*>(out);
    for (int i = blockIdx.x * blockDim.x + threadIdx.x; i < n4;
         i += gridDim.x * blockDim.x) {
        float4 v = o4[i];
        v.x *= scale; v.y *= scale; v.z *= scale; v.w *= scale;
        o4[i] = v;
    }
}

extern "C" void kernel_launch(void* const* d_in, const int* in_sizes, int n_in,
                              void* d_out, int out_size, void* d_ws, size_t ws_size,
                              hipStream_t stream) {
    const float* wf = (const float*)d_in[0];   // waveform  (256, 220500) f32
    const float* wn = (const float*)d_in[1];   // white_noise same shape
    float* out = (float*)d_out;
    unsigned int* mx = (unsigned int*)d_ws;

    pink_init_ws<<<1, 1, 0, stream>>>(mx);

    // 16 channel-groups x 54 time-chunks = 864 waves = 108 blocks of 8 waves
    pink_scan_mix<<<(16 * NCHUNK) / WAVES_PER_BLOCK, 32 * WAVES_PER_BLOCK, 0,
                    stream>>>(wf, wn, out, mx);

    const int n4 = out_size / 4;               // out_size = 56,448,000
    pink_normalize<<<2048, 256, 0, stream>>>(out, mx, n4);
}